// HierarchicalCategoricalLogLikelihoodLossLayer_47991964566133
// MI455X (gfx1250) — compile-verified
//
#include <hip/hip_runtime.h>
#include <math.h>

typedef float v2f __attribute__((ext_vector_type(2)));
typedef float v8f __attribute__((ext_vector_type(8)));

#define Bb 32
#define Tt 512
#define Dd 80
#define Kk 16
#define Pp 64
#define TC 64            // t-chunk size
#define ZP 84            // padded d-pitch for Z chunk [t][d]
#define CP 81            // padded pitch for 80x80 LDS matrices
#define LOG2PI 1.8378770664093453f

// LDS layout (float offsets). Zs aliases Lm (phase-disjoint, barrier-separated).
#define OFF_CS   0        // 80x81 C matrix
#define OFF_LM   6480     // 80x81 L        (also Zs: 64x84 = 5376 <= 6480)
#define OFF_LI   12960    // 80x81 L^-1
#define OFF_SW   19440    // 64 sqrt-gamma
#define OFF_MU   19504    // 80 mu
#define OFF_RED  19584    // 256 reduction scratch
#define SMEM_FLOATS 19840

__device__ __forceinline__ void tile_ij(int t, int& i, int& j) {
  // upper-triangular 5x5 tile enumeration; t==14/15 -> (4,4) (15 is a dummy slot)
  if (t < 5)       { i = 0; j = t; }
  else if (t < 9)  { i = 1; j = t - 4; }
  else if (t < 12) { i = 2; j = t - 7; }
  else if (t < 14) { i = 3; j = t - 9; }
  else             { i = 4; j = 4; }
}

__device__ __forceinline__ float block_reduce256(float v, float* red) {
  const int tid = threadIdx.x;
  red[tid] = v;
  __syncthreads();
  for (int s = 128; s > 0; s >>= 1) {
    if (tid < s) red[tid] += red[tid + s];
    __syncthreads();
  }
  float r = red[0];
  __syncthreads();
  return r;
}

// One workgroup per (b,k):
//   phase 1: C = Z Z^T via WMMA f32 (Z = sqrt(gamma)*(x-mu), 80xT)
//   phase 3: L^-1 by per-column forward substitution
//   phase 4: A = L^-T L^-1 via WMMA; mahal = <A, C> elementwise
__global__ __launch_bounds__(256) void gmm_main(const float* __restrict__ x,
                                                const float* __restrict__ mu_subj,
                                                const float* __restrict__ L_subj,
                                                const float* __restrict__ gam,
                                                const int* __restrict__ sids,
                                                float* __restrict__ accs) {
  extern __shared__ float sm[];
  float* Cs  = sm + OFF_CS;
  float* Lm  = sm + OFF_LM;
  float* Zs  = sm + OFF_LM;   // alias, used only in phase 1
  float* Li  = sm + OFF_LI;
  float* sw  = sm + OFF_SW;
  float* mus = sm + OFF_MU;
  float* red = sm + OFF_RED;

  const int tid = threadIdx.x;
  const int b = blockIdx.x / Kk;
  const int k = blockIdx.x % Kk;
  const int sid = sids[b];
  const float* mu = mu_subj + (sid * Kk + k) * Dd;
  const float* Lg = L_subj + ((size_t)(sid * Kk + k)) * Dd * Dd;

  if (tid < Dd) mus[tid] = mu[tid];

  const int lane = tid & 31;        // wave32
  const int w    = tid >> 5;        // 8 waves
  const int m    = lane & 15;       // A: row M / B: col N
  const int kb   = (lane >> 4) * 2; // K sub-block per lane half
  const int hi   = (lane >> 4) * 8; // C/D fragment M offset per lane half

  int i0, j0, i1, j1;
  tile_ij(w, i0, j0);
  tile_ij(w + 8, i1, j1);
  const int t0i = i0 * 16, t0j = j0 * 16;
  const int t1i = i1 * 16, t1j = j1 * 16;

  v8f c0 = {0.f, 0.f, 0.f, 0.f, 0.f, 0.f, 0.f, 0.f};
  v8f c1 = {0.f, 0.f, 0.f, 0.f, 0.f, 0.f, 0.f, 0.f};
  float swacc = 0.f;  // per-thread partial of sum_t gamma[b,t,k]

  __syncthreads();  // mus visible

  // ---- Phase 1: C += Z_chunk Z_chunk^T over 8 chunks of T ----
  for (int t0 = 0; t0 < Tt; t0 += TC) {
    if (tid < TC) {
      const float g = gam[(b * Tt + t0 + tid) * Kk + k];
      sw[tid] = sqrtf(g);
      swacc += g;
    }
    __syncthreads();
    for (int e = tid; e < Dd * TC; e += 256) {
      int t = e / Dd;
      int d = e - t * Dd;
      Zs[t * ZP + d] = sw[t] * (x[(size_t)(b * Tt + t0 + t) * Dd + d] - mus[d]);
    }
    __syncthreads();
#pragma unroll
    for (int step = 0; step < TC / 4; ++step) {
      const int tt = step * 4 + kb;
      v2f a0, b0, a1, b1;
      a0.x = Zs[tt * ZP + t0i + m];
      a0.y = Zs[(tt + 1) * ZP + t0i + m];
      b0.x = Zs[tt * ZP + t0j + m];
      b0.y = Zs[(tt + 1) * ZP + t0j + m];
      c0 = __builtin_amdgcn_wmma_f32_16x16x4_f32(false, a0, false, b0,
                                                 (short)0, c0, false, false);
      a1.x = Zs[tt * ZP + t1i + m];
      a1.y = Zs[(tt + 1) * ZP + t1i + m];
      b1.x = Zs[tt * ZP + t1j + m];
      b1.y = Zs[(tt + 1) * ZP + t1j + m];
      c1 = __builtin_amdgcn_wmma_f32_16x16x4_f32(false, a1, false, b1,
                                                 (short)0, c1, false, false);
    }
    __syncthreads();
  }

  // ---- Phase 2: fragments -> LDS C (mirror the symmetric half) ----
  {
#pragma unroll
    for (int v = 0; v < 8; ++v) {
      const int r0 = t0i + v + hi, cc0 = t0j + m;
      Cs[r0 * CP + cc0] = c0[v];
      if (t0i != t0j) Cs[cc0 * CP + r0] = c0[v];
      if (w + 8 < 15) {  // slot 15 is a dummy
        const int r1 = t1i + v + hi, cc1 = t1j + m;
        Cs[r1 * CP + cc1] = c1[v];
        if (t1i != t1j) Cs[cc1 * CP + r1] = c1[v];
      }
    }
  }

  // ---- Phase 3: load L, invert by per-column forward substitution ----
  for (int e = tid; e < Dd * Dd; e += 256) {
    int r = e / Dd, cc = e - r * Dd;
    Lm[r * CP + cc] = Lg[e];
  }
  __syncthreads();  // Lm + Cs complete

  if (tid < Dd) {
    const int c = tid;  // each lane owns one column of L^-1 (no cross-lane deps)
    for (int d = 0; d < Dd; ++d) {
      float v;
      if (d < c) {
        v = 0.f;
      } else {
        float s = (d == c) ? 1.f : 0.f;
        for (int j = c; j < d; ++j) s -= Lm[d * CP + j] * Li[j * CP + c];
        v = s / Lm[d * CP + d];
      }
      Li[d * CP + c] = v;
    }
  }
  __syncthreads();

  // ---- Phase 4: A = L^-T L^-1 via WMMA (same tile map), mahal = <A,C> ----
  float part = 0.f;
  {
    v8f a0 = {0.f, 0.f, 0.f, 0.f, 0.f, 0.f, 0.f, 0.f};
    v8f a1 = {0.f, 0.f, 0.f, 0.f, 0.f, 0.f, 0.f, 0.f};
#pragma unroll
    for (int step = 0; step < Dd / 4; ++step) {
      const int dd = step * 4 + kb;
      v2f fa, fb, ga, gb;
      fa.x = Li[dd * CP + t0i + m];
      fa.y = Li[(dd + 1) * CP + t0i + m];
      fb.x = Li[dd * CP + t0j + m];
      fb.y = Li[(dd + 1) * CP + t0j + m];
      a0 = __builtin_amdgcn_wmma_f32_16x16x4_f32(false, fa, false, fb,
                                                 (short)0, a0, false, false);
      ga.x = Li[dd * CP + t1i + m];
      ga.y = Li[(dd + 1) * CP + t1i + m];
      gb.x = Li[dd * CP + t1j + m];
      gb.y = Li[(dd + 1) * CP + t1j + m];
      a1 = __builtin_amdgcn_wmma_f32_16x16x4_f32(false, ga, false, gb,
                                                 (short)0, a1, false, false);
    }
    const float wgt0 = (t0i == t0j) ? 1.f : 2.f;  // symmetric off-diag tiles count twice
#pragma unroll
    for (int v = 0; v < 8; ++v)
      part += wgt0 * a0[v] * Cs[(t0i + v + hi) * CP + (t0j + m)];
    if (w + 8 < 15) {  // skip dummy slot's contribution
      const float wgt1 = (t1i == t1j) ? 1.f : 2.f;
#pragma unroll
      for (int v = 0; v < 8; ++v)
        part += wgt1 * a1[v] * Cs[(t1i + v + hi) * CP + (t1j + m)];
    }
  }

  float dlog = (tid < Dd) ? logf(Lm[tid * CP + tid]) : 0.f;

  const float mahalw = block_reduce256(part, red);
  const float sdiag  = block_reduce256(dlog, red);
  const float sumw   = block_reduce256(swacc, red);

  if (tid == 0) {
    const float logdet = 2.f * sdiag;
    const float gl = -0.5f * ((float)Dd * LOG2PI + logdet) * sumw - 0.5f * mahalw;
    atomicAdd(&accs[0], -gl / (float)(Bb * Tt));
  }
}

__global__ void mask_kernel(const int* __restrict__ sids, float* __restrict__ accs,
                            float* __restrict__ mask) {
  __shared__ float red[64];
  const int p = threadIdx.x;
  float mval = 0.f;
  for (int bb = 0; bb < Bb; ++bb)
    if (sids[bb] == p) mval = 1.f;
  mask[p] = mval;
  red[p] = mval;
  __syncthreads();
  for (int s = 32; s > 0; s >>= 1) {
    if (p < s) red[p] += red[p + s];
    __syncthreads();
  }
  if (p == 0) {
    accs[0] = 0.f;    // nll accumulator
    accs[1] = 0.f;    // reg accumulator
    accs[2] = red[0]; // S = #subjects present
  }
}

__global__ __launch_bounds__(256) void reg_kernel(const float* __restrict__ L_subj,
                                                  const float* __restrict__ L_pop,
                                                  const float* __restrict__ mu_subj,
                                                  const float* __restrict__ mu_pop,
                                                  const float* __restrict__ mask,
                                                  float* __restrict__ accs) {
  __shared__ float red[256];
  const long long N1 = (long long)Pp * Kk * Dd * Dd;  // L terms
  const long long N2 = (long long)Pp * Kk * Dd;       // mu terms
  const long long N = N1 + N2;
  float acc = 0.f;
  for (long long i = (long long)blockIdx.x * 256 + threadIdx.x; i < N;
       i += (long long)gridDim.x * 256) {
    float d;
    int p;
    if (i < N1) {
      const int ii = (int)i;
      p = ii / (Kk * Dd * Dd);
      d = L_subj[ii] - L_pop[ii % (Kk * Dd * Dd)];
    } else {
      const int ii = (int)(i - N1);
      p = ii / (Kk * Dd);
      d = mu_subj[ii] - mu_pop[ii % (Kk * Dd)];
    }
    acc += mask[p] * d * d;
  }
  red[threadIdx.x] = acc;
  __syncthreads();
  for (int s = 128; s > 0; s >>= 1) {
    if (threadIdx.x < s) red[threadIdx.x] += red[threadIdx.x + s];
    __syncthreads();
  }
  if (threadIdx.x == 0) atomicAdd(&accs[1], red[0]);
}

__global__ void final_kernel(const float* __restrict__ accs, float* __restrict__ out) {
  // scale = S/64; LAMBDA_MU/2 = LAMBDA_L/2 = 0.05 so the two regs fold together
  out[0] = accs[0] + (accs[2] / 64.0f) * 0.05f * accs[1];
}

extern "C" void kernel_launch(void* const* d_in, const int* in_sizes, int n_in,
                              void* d_out, int out_size, void* d_ws, size_t ws_size,
                              hipStream_t stream) {
  const float* x       = (const float*)d_in[0];
  const float* mu_pop  = (const float*)d_in[1];
  const float* L_pop   = (const float*)d_in[2];
  const float* mu_subj = (const float*)d_in[3];
  const float* L_subj  = (const float*)d_in[4];
  const float* gam     = (const float*)d_in[5];
  const int*   sids    = (const int*)d_in[6];
  float* out  = (float*)d_out;
  float* accs = (float*)d_ws;       // [0]=nll, [1]=reg, [2]=S
  float* mask = accs + 16;          // 64 floats

  mask_kernel<<<dim3(1), dim3(64), 0, stream>>>(sids, accs, mask);
  gmm_main<<<dim3(Bb * Kk), dim3(256), SMEM_FLOATS * sizeof(float), stream>>>(
      x, mu_subj, L_subj, gam, sids, accs);
  reg_kernel<<<dim3(1024), dim3(256), 0, stream>>>(L_subj, L_pop, mu_subj, mu_pop,
                                                   mask, accs);
  final_kernel<<<dim3(1), dim3(1), 0, stream>>>(accs, out);
}